// Encoder_46806553591978
// MI455X (gfx1250) — compile-verified
//
#include <hip/hip_runtime.h>
#include <hip/hip_bf16.h>

typedef __attribute__((ext_vector_type(2))) float v2f;
typedef __attribute__((ext_vector_type(8))) float v8f;

#define DCOL 768            // embedding width
#define GROUP_WORDS 16      // words per block (one WMMA M-tile)
#define GROUP_ROWS 32       // subword rows per block
#define COL_TILES (DCOL / 16)   // 48 column tiles of 16

// One block handles 16 words (32 subword rows).
// Phase 1: scores s = E . w + b for the 32 rows, then 2-way softmax -> p[16][2].
// Phase 2: out[16x768] = P(16x32, block-diagonal) x E(32x768) via chained
//          V_WMMA_F32_16X16X4_F32 (8 K-chunks of 4), full f32 precision.
__global__ __launch_bounds__(256) void word_pool_wmma(
    const float* __restrict__ emb,      // [16384, 768]
    const int*   __restrict__ off,      // [8192, 2] (start, end), end = start+2
    const float* __restrict__ attn_w,   // [768]
    const float* __restrict__ attn_b,   // [1]
    float*       __restrict__ out,      // [8192, 768]
    int n_words)
{
    __shared__ float s_w[DCOL];
    __shared__ float s_partial[GROUP_ROWS][8];
    __shared__ float s_p[GROUP_WORDS][2];

    const int tid       = threadIdx.x;
    const int g         = blockIdx.x;
    const int word_base = g * GROUP_WORDS;

    // Stage attn_w into LDS (reused by 32 row-dot-products).
    for (int i = tid; i < DCOL; i += 256) s_w[i] = attn_w[i];
    __syncthreads();

    // ---- Phase 1: scores for the 32 subword rows of this group ----
    {
        const int r   = tid >> 3;   // 0..31 local subword
        const int sub = tid & 7;    // 8 threads per row
        const int w   = word_base + (r >> 1);
        const int row = off[2 * w] + (r & 1);   // honor offsets (contiguous 2-span)
        const float* rowp = emb + (size_t)row * DCOL;
        float acc = 0.f;
        #pragma unroll 4
        for (int j = sub * 4; j < DCOL; j += 32) {
            float4 e  = *(const float4*)(rowp + j);
            float4 ww = *(const float4*)(s_w + j);
            acc += e.x * ww.x + e.y * ww.y + e.z * ww.z + e.w * ww.w;
        }
        s_partial[r][sub] = acc;
    }
    __syncthreads();
    if (tid < GROUP_ROWS) {
        float s = attn_b[0];
        #pragma unroll
        for (int k = 0; k < 8; ++k) s += s_partial[tid][k];
        s_partial[tid][0] = s;      // reuse slot 0 as the score
    }
    __syncthreads();
    if (tid < GROUP_WORDS) {
        float s0 = s_partial[2 * tid + 0][0];
        float s1 = s_partial[2 * tid + 1][0];
        float m  = fmaxf(s0, s1);
        float e0 = expf(s0 - m);
        float e1 = expf(s1 - m);
        float inv = 1.f / (e0 + e1);
        s_p[tid][0] = e0 * inv;
        s_p[tid][1] = e1 * inv;
    }
    __syncthreads();

    // ---- Phase 2: out = P x E with f32 WMMA ----
    const int lane = tid & 31;      // wave32
    const int wv   = tid >> 5;      // 8 waves per block
    const int n    = lane & 15;     // A-row m / B-col n / D-col n
    const int kh   = lane >> 4;     // K half within a 4-chunk

    // Build the 8 A chunks (16x4 each). Word m's (p0,p1) lives at K-chunk
    // kc = m>>1, half kh = m&1, VGPR0/VGPR1. Everything else is zero.
    v2f A[8];
    #pragma unroll
    for (int kc = 0; kc < 8; ++kc) {
        bool nz = (n == 2 * kc + kh);
        A[kc].x = nz ? s_p[n][0] : 0.f;
        A[kc].y = nz ? s_p[n][1] : 0.f;
    }

    for (int t = wv; t < COL_TILES; t += 8) {
        const int c0 = t * 16;
        v8f acc = {};
        #pragma unroll
        for (int kc = 0; kc < 8; ++kc) {
            // B chunk rows: subwords of word (word_base + 2*kc + kh)
            const int bw = word_base + 2 * kc + kh;
            const int r0 = off[2 * bw];
            v2f B;
            B.x = emb[(size_t)r0 * DCOL + c0 + n];
            B.y = emb[(size_t)(r0 + 1) * DCOL + c0 + n];
            acc = __builtin_amdgcn_wmma_f32_16x16x4_f32(
                false, A[kc], false, B, (short)0, acc, false, false);
        }
        // D layout: VGPR r, lane-half kh -> output row r + 8*kh
        #pragma unroll
        for (int r = 0; r < 8; ++r) {
            out[(size_t)(word_base + r + 8 * kh) * DCOL + c0 + n] = acc[r];
        }
    }
}

extern "C" void kernel_launch(void* const* d_in, const int* in_sizes, int n_in,
                              void* d_out, int out_size, void* d_ws, size_t ws_size,
                              hipStream_t stream) {
    const float* emb    = (const float*)d_in[0];
    const int*   off    = (const int*)d_in[1];
    const float* attn_w = (const float*)d_in[2];
    const float* attn_b = (const float*)d_in[3];
    float*       out    = (float*)d_out;

    const int n_words = in_sizes[1] / 2;           // 8192
    const int nblocks = n_words / GROUP_WORDS;     // 512

    hipLaunchKernelGGL(word_pool_wmma, dim3(nblocks), dim3(256), 0, stream,
                       emb, off, attn_w, attn_b, out, n_words);
}